// improvedDeformableLocalCrossAttention_14826227106563
// MI455X (gfx1250) — compile-verified
//
#include <hip/hip_runtime.h>
#include <math.h>

// ---------------------------------------------------------------------------
// Problem constants (from reference): B=4, N=1024, C=384, H=8, G=2, K=16
// ---------------------------------------------------------------------------
constexpr int Bc  = 4;
constexpr int Nc  = 1024;
constexpr int Cc  = 384;
constexpr int Hc  = 8;      // heads
constexpr int HDc = 48;     // head dim
constexpr int Gc  = 2;      // groups
constexpr int GCc = 192;    // group channels
constexpr int Kc  = 16;     // kNN neighbors

constexpr int BN   = Bc * Nc;        // 4096 tokens
constexpr int NK   = Nc * Kc;        // 16384
constexpr int BGNK = Bc * Gc * NK;   // 131072 shifted points
constexpr float BIGF = 3.0e38f;

typedef __attribute__((ext_vector_type(2))) float v2f;
typedef __attribute__((ext_vector_type(8))) float v8f;

// fp32 WMMA: D(16x16) = A(16x4) * B(4x16) + C   — keeps exact fp32 numerics
__device__ __forceinline__ v8f wmma4(v2f a, v2f b, v8f c) {
  return __builtin_amdgcn_wmma_f32_16x16x4_f32(
      /*neg_a=*/false, a, /*neg_b=*/false, b,
      /*c_mod=*/(short)0, c, /*reuse_a=*/false, /*reuse_b=*/false);
}

__device__ __forceinline__ bool dless(float d, int i, float dj, int ij) {
  // strict-weak order with stable index tie-break (matches jax.lax.top_k).
  // sentinel index -1 becomes huge as unsigned, so it always loses ties.
  return (d < dj) || (d == dj && (unsigned)i < (unsigned)ij);
}

// ---------------------------------------------------------------------------
// Kernel 1: brute-force kNN in 3D, top-K=16 smallest squared distances
// One 256-thread block per query token.
// ---------------------------------------------------------------------------
__global__ __launch_bounds__(256)
void knn_kernel(const float* __restrict__ qpos, int* __restrict__ idx) {
  __shared__ float dist[Nc];
  __shared__ float rd[256];
  __shared__ int   ri[256];
  const int qid = blockIdx.x;           // [0, B*N)
  const int b = qid >> 10;
  const int n = qid & 1023;
  const int t = threadIdx.x;

  const float px = qpos[(b * Nc + n) * 3 + 0];
  const float py = qpos[(b * Nc + n) * 3 + 1];
  const float pz = qpos[(b * Nc + n) * 3 + 2];

  for (int m = t; m < Nc; m += 256) {
    float dx = px - qpos[(b * Nc + m) * 3 + 0];
    float dy = py - qpos[(b * Nc + m) * 3 + 1];
    float dz = pz - qpos[(b * Nc + m) * 3 + 2];
    dist[m] = dx * dx + dy * dy + dz * dz;
  }
  __syncthreads();

  for (int r = 0; r < Kc; ++r) {
    float bd = BIGF; int bi = -1;
    for (int m = t; m < Nc; m += 256) {
      float d = dist[m];
      if (dless(d, m, bd, bi)) { bd = d; bi = m; }
    }
    rd[t] = bd; ri[t] = bi;
    __syncthreads();
    for (int s = 128; s > 0; s >>= 1) {
      if (t < s) {
        float d2 = rd[t + s]; int i2 = ri[t + s];
        if (dless(d2, i2, rd[t], ri[t])) { rd[t] = d2; ri[t] = i2; }
      }
      __syncthreads();
    }
    if (t == 0) {
      idx[qid * Kc + r] = ri[0];
      dist[ri[0]] = BIGF;               // exclude from next round
    }
    __syncthreads();
  }
}

// ---------------------------------------------------------------------------
// Kernel 2: qp = q @ Wq ; v_off = q @ Wvoff   (rows = 4096, C=384)
// 8 waves / block, each wave owns one 16x16 output tile, 96 WMMA k-steps.
// grid = (tiles/8, 2)  with tiles = (4096/16)*(384/16) = 6144
// ---------------------------------------------------------------------------
__global__ __launch_bounds__(256)
void gemm_qp_voff_kernel(const float* __restrict__ Ain,
                         const float* __restrict__ Wq,
                         const float* __restrict__ Wvoff,
                         float* __restrict__ qp, float* __restrict__ voff) {
  const int wave = threadIdx.x >> 5;
  const int lane = threadIdx.x & 31;
  const int half = lane >> 4;
  const int r    = lane & 15;

  const int tile = blockIdx.x * 8 + wave;      // < 6144
  const int mt = tile / (Cc / 16);
  const int nt = tile % (Cc / 16);
  const float* __restrict__ W = (blockIdx.y == 0) ? Wq : Wvoff;
  float* __restrict__ O       = (blockIdx.y == 0) ? qp : voff;

  v8f c = {};
  const int arow = (mt * 16 + r) * Cc;
  for (int k0 = 0; k0 < Cc; k0 += 4) {
    if ((k0 & 31) == 0 && k0 + 32 < Cc)
      __builtin_prefetch(&W[(k0 + 32) * Cc + nt * 16], 0, 1);
    v2f a, bb;
    a.x  = Ain[arow + k0 + 2 * half + 0];
    a.y  = Ain[arow + k0 + 2 * half + 1];
    bb.x = W[(k0 + 2 * half + 0) * Cc + nt * 16 + r];
    bb.y = W[(k0 + 2 * half + 1) * Cc + nt * 16 + r];
    c = wmma4(a, bb, c);
  }
  for (int v = 0; v < 8; ++v)
    O[(mt * 16 + v + 8 * half) * Cc + nt * 16 + r] = c[v];
}

// ---------------------------------------------------------------------------
// Kernel 3: fused offset MLP per (bg, n):
//   A[16x384] = concat(v_off[gathered, group g], qp[n, group g])
//   h = A @ W_off1 + b_off1 ; LayerNorm ; exact GELU ; tanh(h @ W_off2)
//   shift_pos = local_v_pos + offset * 0.5*(max-min)
// ---------------------------------------------------------------------------
__global__ __launch_bounds__(256)
void offset_mlp_kernel(const float* __restrict__ qpos,
                       const float* __restrict__ qp,
                       const float* __restrict__ voff,
                       const int*   __restrict__ idx,
                       const float* __restrict__ W1, const float* __restrict__ b1,
                       const float* __restrict__ lng, const float* __restrict__ lnb,
                       const float* __restrict__ W2,
                       float* __restrict__ shift_pos) {
  __shared__ float As[Kc][Cc];     // 24 KB
  __shared__ float Hs[Kc][Cc];     // 24 KB
  __shared__ int   id_s[Kc];
  __shared__ float mu_s[Kc], rs_s[Kc];
  __shared__ float off_s[Kc][3];
  __shared__ float lp[Kc][3];
  __shared__ float sc_s[3];

  const int bgn = blockIdx.x;      // [0, B*G*N)
  const int bg  = bgn / Nc;
  const int n   = bgn % Nc;
  const int b   = bg / Gc;
  const int g   = bg % Gc;
  const int t   = threadIdx.x;

  if (t < Kc) id_s[t] = idx[(b * Nc + n) * Kc + t];
  __syncthreads();

  for (int e = t; e < Kc * Cc; e += 256) {
    int k = e / Cc, ch = e % Cc;
    float v;
    if (ch < GCc) v = voff[(b * Nc + id_s[k]) * Cc + g * GCc + ch];
    else          v = qp[(b * Nc + n) * Cc + g * GCc + (ch - GCc)];
    As[k][ch] = v;
  }
  __syncthreads();

  // h = A @ W_off1 + b_off1  (24 tiles of 16x16, 3 per wave)
  const int wave = t >> 5, lane = t & 31, half = lane >> 4, r = lane & 15;
  for (int nt = wave; nt < Cc / 16; nt += 8) {
    v8f c = {};
    for (int k0 = 0; k0 < Cc; k0 += 4) {
      v2f a, bb;
      a.x  = As[r][k0 + 2 * half + 0];
      a.y  = As[r][k0 + 2 * half + 1];
      bb.x = W1[(k0 + 2 * half + 0) * Cc + nt * 16 + r];
      bb.y = W1[(k0 + 2 * half + 1) * Cc + nt * 16 + r];
      c = wmma4(a, bb, c);
    }
    const float bias = b1[nt * 16 + r];
    for (int v = 0; v < 8; ++v)
      Hs[v + 8 * half][nt * 16 + r] = c[v] + bias;
  }
  __syncthreads();

  // LayerNorm stats per row (K=16 rows)
  if (t < Kc) {
    float s = 0.f;
    for (int c = 0; c < Cc; ++c) s += Hs[t][c];
    float mu = s * (1.0f / Cc);
    float vs = 0.f;
    for (int c = 0; c < Cc; ++c) { float d = Hs[t][c] - mu; vs += d * d; }
    mu_s[t] = mu;
    rs_s[t] = rsqrtf(vs * (1.0f / Cc) + 1e-5f);
  }
  __syncthreads();

  // normalize + affine + exact GELU
  for (int e = t; e < Kc * Cc; e += 256) {
    int k = e / Cc, ch = e % Cc;
    float x = (Hs[k][ch] - mu_s[k]) * rs_s[k] * lng[ch] + lnb[ch];
    x = 0.5f * x * (1.0f + erff(x * 0.70710678118654752f));
    Hs[k][ch] = x;
  }
  __syncthreads();

  // offset = tanh(H @ W_off2) and local positions
  if (t < Kc * 3) {
    int k = t / 3, d = t % 3;
    float s = 0.f;
    for (int c = 0; c < Cc; ++c) s += Hs[k][c] * W2[c * 3 + d];
    off_s[k][d] = tanhf(s);
  }
  if (t >= 64 && t < 64 + Kc * 3) {
    int e = t - 64; int k = e / 3, d = e % 3;
    lp[k][d] = qpos[(b * Nc + id_s[k]) * 3 + d];
  }
  __syncthreads();
  if (t < 3) {
    float mx = -BIGF, mn = BIGF;
    for (int k = 0; k < Kc; ++k) { float v = lp[k][t]; mx = fmaxf(mx, v); mn = fminf(mn, v); }
    sc_s[t] = (mx - mn) * 0.5f;
  }
  __syncthreads();
  if (t < Kc * 3) {
    int k = t / 3, d = t % 3;
    shift_pos[((long)bg * NK + n * Kc + k) * 3 + d] = lp[k][d] + off_s[k][d] * sc_s[d];
  }
}

// ---------------------------------------------------------------------------
// Kernel 4: three_nn over 1024 candidates per shifted point + IDW weights.
// One wave per point, 8 points per 256-thread block.
// ---------------------------------------------------------------------------
__global__ __launch_bounds__(256)
void three_nn_kernel(const float* __restrict__ qpos,
                     const float* __restrict__ shift_pos,
                     int* __restrict__ idx3, float* __restrict__ w3) {
  __shared__ float cd[8][96];
  __shared__ int   ci[8][96];
  const int wave = threadIdx.x >> 5, lane = threadIdx.x & 31;
  const long p = (long)blockIdx.x * 8 + wave;     // < 131072
  const int bg = (int)(p / NK);
  const int b  = bg / Gc;

  const float sx = shift_pos[p * 3 + 0];
  const float sy = shift_pos[p * 3 + 1];
  const float sz = shift_pos[p * 3 + 2];

  float d0 = BIGF, d1 = BIGF, d2 = BIGF;
  int   i0 = -1,  i1 = -1,  i2 = -1;
  for (int m = lane; m < Nc; m += 32) {
    float dx = sx - qpos[(b * Nc + m) * 3 + 0];
    float dy = sy - qpos[(b * Nc + m) * 3 + 1];
    float dz = sz - qpos[(b * Nc + m) * 3 + 2];
    float d = dx * dx + dy * dy + dz * dz;
    if (dless(d, m, d0, i0))      { d2 = d1; i2 = i1; d1 = d0; i1 = i0; d0 = d; i0 = m; }
    else if (dless(d, m, d1, i1)) { d2 = d1; i2 = i1; d1 = d;  i1 = m; }
    else if (dless(d, m, d2, i2)) { d2 = d;  i2 = m; }
  }
  cd[wave][lane * 3 + 0] = d0; ci[wave][lane * 3 + 0] = i0;
  cd[wave][lane * 3 + 1] = d1; ci[wave][lane * 3 + 1] = i1;
  cd[wave][lane * 3 + 2] = d2; ci[wave][lane * 3 + 2] = i2;
  __syncthreads();

  if (lane == 0) {
    float bd[3] = {BIGF, BIGF, BIGF};
    int   bi[3] = {-1, -1, -1};
    for (int e = 0; e < 96; ++e) {
      float d = cd[wave][e]; int i = ci[wave][e];
      if (dless(d, i, bd[0], bi[0]))      { bd[2]=bd[1]; bi[2]=bi[1]; bd[1]=bd[0]; bi[1]=bi[0]; bd[0]=d; bi[0]=i; }
      else if (dless(d, i, bd[1], bi[1])) { bd[2]=bd[1]; bi[2]=bi[1]; bd[1]=d; bi[1]=i; }
      else if (dless(d, i, bd[2], bi[2])) { bd[2]=d; bi[2]=i; }
    }
    float w[3], s = 0.f;
    for (int i = 0; i < 3; ++i) {
      float dist = sqrtf(fmaxf(bd[i], 0.0f));
      w[i] = 1.0f / (dist + 1e-8f);
      s += w[i];
    }
    const float inv = 1.0f / s;
    for (int i = 0; i < 3; ++i) {
      idx3[p * 3 + i] = bi[i];
      w3[p * 3 + i]   = w[i] * inv;
    }
  }
}

// ---------------------------------------------------------------------------
// Kernel 5: per (b,n) — interpolate 16x384 panel, WMMA @Wk, softmax attn,
// WMMA @Wv (reusing LDS), weighted sum -> attn_out.  LDS ~49 KB.
// ---------------------------------------------------------------------------
__global__ __launch_bounds__(256)
void attn_kernel(const float* __restrict__ q,
                 const float* __restrict__ qp,
                 const int*   __restrict__ idx3,
                 const float* __restrict__ w3,
                 const float* __restrict__ Wk,
                 const float* __restrict__ Wv,
                 float* __restrict__ attn_out) {
  __shared__ float I[Kc][Cc];      // interp panel, 24 KB
  __shared__ float KV[Kc][Cc];     // Kf then Vf,   24 KB
  __shared__ float att[Hc][Kc];

  const int bn = blockIdx.x;       // [0, 4096)
  const int b  = bn >> 10;
  const int n  = bn & 1023;
  const int t  = threadIdx.x;

  // three_interpolate gather into I
  for (int e = t; e < Kc * Cc; e += 256) {
    int k = e / Cc, ch = e % Cc;
    int g = ch / GCc;
    long p = ((long)(b * Gc + g)) * NK + n * Kc + k;
    float s = 0.f;
    for (int i = 0; i < 3; ++i) {
      int m = idx3[p * 3 + i];
      s += w3[p * 3 + i] * q[(b * Nc + m) * Cc + ch];
    }
    I[k][ch] = s;
  }
  __syncthreads();

  const int wave = t >> 5, lane = t & 31, half = lane >> 4, r = lane & 15;

  // Kf = I @ Wk
  for (int nt = wave; nt < Cc / 16; nt += 8) {
    v8f c = {};
    for (int k0 = 0; k0 < Cc; k0 += 4) {
      v2f a, bb;
      a.x  = I[r][k0 + 2 * half + 0];
      a.y  = I[r][k0 + 2 * half + 1];
      bb.x = Wk[(k0 + 2 * half + 0) * Cc + nt * 16 + r];
      bb.y = Wk[(k0 + 2 * half + 1) * Cc + nt * 16 + r];
      c = wmma4(a, bb, c);
    }
    for (int v = 0; v < 8; ++v)
      KV[v + 8 * half][nt * 16 + r] = c[v];
  }
  __syncthreads();

  // attention logits + softmax (K=16 keys per head)
  const float scale = 0.144337567297406441f;    // 48^-0.5
  if (t < Hc * Kc) {
    int h = t >> 4, k = t & 15;
    float s = 0.f;
    for (int c = 0; c < HDc; ++c)
      s += qp[(b * Nc + n) * Cc + h * HDc + c] * KV[k][h * HDc + c];
    att[h][k] = s * scale;
  }
  __syncthreads();
  if (t < Hc) {
    float mx = -BIGF;
    for (int k = 0; k < Kc; ++k) mx = fmaxf(mx, att[t][k]);
    float s = 0.f;
    for (int k = 0; k < Kc; ++k) { float e = expf(att[t][k] - mx); att[t][k] = e; s += e; }
    float inv = 1.0f / s;
    for (int k = 0; k < Kc; ++k) att[t][k] *= inv;
  }
  __syncthreads();

  // Vf = I @ Wv  (reuse KV buffer)
  for (int nt = wave; nt < Cc / 16; nt += 8) {
    v8f c = {};
    for (int k0 = 0; k0 < Cc; k0 += 4) {
      v2f a, bb;
      a.x  = I[r][k0 + 2 * half + 0];
      a.y  = I[r][k0 + 2 * half + 1];
      bb.x = Wv[(k0 + 2 * half + 0) * Cc + nt * 16 + r];
      bb.y = Wv[(k0 + 2 * half + 1) * Cc + nt * 16 + r];
      c = wmma4(a, bb, c);
    }
    for (int v = 0; v < 8; ++v)
      KV[v + 8 * half][nt * 16 + r] = c[v];
  }
  __syncthreads();

  // out[ch] = sum_k att[h][k] * Vf[k][ch]
  for (int ch = t; ch < Cc; ch += 256) {
    int h = ch / HDc;
    float s = 0.f;
    for (int k = 0; k < Kc; ++k) s += att[h][k] * KV[k][ch];
    attn_out[(b * Nc + n) * Cc + ch] = s;
  }
}

// ---------------------------------------------------------------------------
// Kernel 6: d_out = attn_out @ Wp + bp   (WMMA, same tiling as kernel 2)
// ---------------------------------------------------------------------------
__global__ __launch_bounds__(256)
void final_gemm_kernel(const float* __restrict__ Ain,
                       const float* __restrict__ Wp,
                       const float* __restrict__ bp,
                       float* __restrict__ out) {
  const int wave = threadIdx.x >> 5;
  const int lane = threadIdx.x & 31;
  const int half = lane >> 4;
  const int r    = lane & 15;

  const int tile = blockIdx.x * 8 + wave;
  const int mt = tile / (Cc / 16);
  const int nt = tile % (Cc / 16);

  v8f c = {};
  const int arow = (mt * 16 + r) * Cc;
  for (int k0 = 0; k0 < Cc; k0 += 4) {
    if ((k0 & 31) == 0 && k0 + 32 < Cc)
      __builtin_prefetch(&Wp[(k0 + 32) * Cc + nt * 16], 0, 1);
    v2f a, bb;
    a.x  = Ain[arow + k0 + 2 * half + 0];
    a.y  = Ain[arow + k0 + 2 * half + 1];
    bb.x = Wp[(k0 + 2 * half + 0) * Cc + nt * 16 + r];
    bb.y = Wp[(k0 + 2 * half + 1) * Cc + nt * 16 + r];
    c = wmma4(a, bb, c);
  }
  const float bias = bp[nt * 16 + r];
  for (int v = 0; v < 8; ++v)
    out[(mt * 16 + v + 8 * half) * Cc + nt * 16 + r] = c[v] + bias;
}

// ---------------------------------------------------------------------------
// Host launch
// ---------------------------------------------------------------------------
extern "C" void kernel_launch(void* const* d_in, const int* in_sizes, int n_in,
                              void* d_out, int out_size, void* d_ws, size_t ws_size,
                              hipStream_t stream) {
  (void)in_sizes; (void)n_in; (void)out_size; (void)ws_size;
  const float* q      = (const float*)d_in[0];
  const float* q_pos  = (const float*)d_in[1];
  const float* Wq     = (const float*)d_in[2];
  const float* Wk     = (const float*)d_in[3];
  const float* Wv     = (const float*)d_in[4];
  const float* Wvoff  = (const float*)d_in[5];
  const float* W_off1 = (const float*)d_in[6];
  const float* b_off1 = (const float*)d_in[7];
  const float* ln_g   = (const float*)d_in[8];
  const float* ln_b   = (const float*)d_in[9];
  const float* W_off2 = (const float*)d_in[10];
  const float* Wp     = (const float*)d_in[11];
  const float* bp     = (const float*)d_in[12];

  // workspace layout (floats), total ~23.9 MB
  float* fws  = (float*)d_ws;
  int*   idx  = (int*)fws;                       // BN*K           = 65536
  float* qp   = fws + BN * Kc;                   // BN*C           = 1572864
  float* voff = qp + (size_t)BN * Cc;            // BN*C
  float* shp  = voff + (size_t)BN * Cc;          // BGNK*3         = 393216
  int*   idx3 = (int*)(shp + (size_t)BGNK * 3);  // BGNK*3
  float* w3   = (float*)(idx3 + (size_t)BGNK * 3);
  float* aout = w3 + (size_t)BGNK * 3;           // BN*C

  // 1) kNN
  knn_kernel<<<BN, 256, 0, stream>>>(q_pos, idx);

  // 2) qp = q@Wq, v_off = q@Wvoff   (6144 tiles, 8 per block)
  gemm_qp_voff_kernel<<<dim3((BN / 16) * (Cc / 16) / 8, 2), 256, 0, stream>>>(
      q, Wq, Wvoff, qp, voff);

  // 3) fused offset MLP -> shift_pos   (B*G*N blocks)
  offset_mlp_kernel<<<Bc * Gc * Nc, 256, 0, stream>>>(
      q_pos, qp, voff, idx, W_off1, b_off1, ln_g, ln_b, W_off2, shp);

  // 4) three_nn + inverse-distance weights
  three_nn_kernel<<<BGNK / 8, 256, 0, stream>>>(q_pos, shp, idx3, w3);

  // 5) interpolate + local cross-attention
  attn_kernel<<<BN, 256, 0, stream>>>(q, qp, idx3, w3, Wk, Wv, aout);

  // 6) final projection
  final_gemm_kernel<<<(BN / 16) * (Cc / 16) / 8, 256, 0, stream>>>(
      aout, Wp, bp, (float*)d_out);
}